// MultiHeadAttention_88252987998311
// MI455X (gfx1250) — compile-verified
//
#include <hip/hip_runtime.h>
#include <hip/hip_bf16.h>
#include <math.h>

typedef __attribute__((ext_vector_type(2))) float v2f;
typedef __attribute__((ext_vector_type(8))) float v8f;

#define D_MODEL   512
#define NHEAD     8
#define HEAD_DIM  64
#define T_SEQ     2048
#define N_BATCH   4
#define WIN_L     127
#define WIN_R     128
// ln(10000)/32  (RoPE inv_freq exponent step, ROTARY_DIM=64 -> half=32)
#define ROPE_LN_STEP 0.28782313662425572f

// fp32 WMMA: D(16x16,f32) = A(16x4,f32) * B(4x16,f32) + C
// A frag: lane l holds A[l&15, 2*(l>>4)] , A[l&15, 2*(l>>4)+1]
// B frag: lane l holds B[2*(l>>4), l&15] , B[2*(l>>4)+1, l&15]
// C/D   : VGPR r, lanes0-15 -> (M=r, N=lane), lanes16-31 -> (M=8+r, N=lane-16)
static __device__ __forceinline__ v8f wmma_f32(v2f a, v2f b, v8f c) {
  return __builtin_amdgcn_wmma_f32_16x16x4_f32(
      /*neg_a=*/false, a, /*neg_b=*/false, b,
      /*c_mod=*/(short)0, c, /*reuse_a=*/false, /*reuse_b=*/false);
}

// ---------------------------------------------------------------------------
// Kernel 1: qkv = x @ Wqkv^T, fused RoPE on q,k; scatter to [n,h,t,d] buffers.
// One wave computes a 16(M) x 64(E) tile == one head's full dim for 16 tokens.
// ---------------------------------------------------------------------------
__global__ __launch_bounds__(256) void qkv_rope_kernel(
    const float* __restrict__ x,     // [N*T, 512]
    const float* __restrict__ w,     // [1536, 512]
    float* __restrict__ qws, float* __restrict__ kws, float* __restrict__ vws)
{
  const int lane = threadIdx.x & 31;
  const int wid  = (int)((blockIdx.x * blockDim.x + threadIdx.x) >> 5);
  const int nb   = wid % 24;          // 64-wide column group -> e0
  const int mt   = wid / 24;          // 16-row tile
  const int m0   = mt * 16;
  const int e0   = nb * 64;
  const int row  = lane & 15;
  const int hi   = lane >> 4;

  v8f acc[4] = {};

  const float* xrow = x + (size_t)(m0 + row) * D_MODEL;
  const float* wrow0 = w + (size_t)(e0 + row) * D_MODEL;

  for (int k0 = 0; k0 < D_MODEL; k0 += 64) {
    // prefetch next 64-float K-block (global_prefetch_b8); L2-resident data,
    // this hides L2 hit latency behind the 16-deep WMMA chain below
    if (k0 + 64 < D_MODEL) {
      __builtin_prefetch(xrow + k0 + 64 + 2 * hi, 0, 1);
      __builtin_prefetch(wrow0 + k0 + 64 + 2 * hi, 0, 1);
    }
#pragma unroll
    for (int kk = 0; kk < 64; kk += 4) {
      const int k = k0 + kk;
      v2f a = *(const v2f*)(xrow + k + 2 * hi);
#pragma unroll
      for (int s = 0; s < 4; ++s) {
        // B[k, col] = W[e0 + col, k]; col = lane&15; contiguous in k -> b64
        v2f b = *(const v2f*)(wrow0 + (size_t)s * 16 * D_MODEL + k + 2 * hi);
        acc[s] = wmma_f32(a, b, acc[s]);
      }
    }
  }

  const int qkv   = e0 / D_MODEL;            // 0=q 1=k 2=v
  const int h     = (e0 % D_MODEL) / HEAD_DIM;
  const int n     = m0 / T_SEQ;
  const int tbase = m0 % T_SEQ;

  if (qkv < 2) {
    // RoPE: pair (d, d+32) == accumulator subtiles (s, s+2), same lane/VGPR.
#pragma unroll
    for (int s = 0; s < 2; ++s) {
      const int d = s * 16 + row;            // 0..31 = pair index
      const float invf = __expf(-(float)d * ROPE_LN_STEP);
#pragma unroll
      for (int r = 0; r < 8; ++r) {
        const int t = tbase + r + 8 * hi;
        float sn, cs;
        __sincosf((float)t * invf, &sn, &cs);
        const float x1 = acc[s][r], x2 = acc[s + 2][r];
        acc[s][r]     = x1 * cs - x2 * sn;
        acc[s + 2][r] = x2 * cs + x1 * sn;
      }
    }
  }

  float* dst = (qkv == 0) ? qws : ((qkv == 1) ? kws : vws);
  const size_t base = ((size_t)(n * NHEAD + h)) * T_SEQ * HEAD_DIM;
#pragma unroll
  for (int s = 0; s < 4; ++s) {
    const int d = s * 16 + row;
#pragma unroll
    for (int r = 0; r < 8; ++r) {
      const int t = tbase + r + 8 * hi;
      dst[base + (size_t)t * HEAD_DIM + d] = acc[s][r];
    }
  }
}

// ---------------------------------------------------------------------------
// Kernel 2: windowed attention with online softmax.
// One wave handles (n, h, 16-query tile); 17 key tiles cover the +-window.
// ---------------------------------------------------------------------------
__global__ __launch_bounds__(256) void attn_kernel(
    const float* __restrict__ qws, const float* __restrict__ kws,
    const float* __restrict__ vws, float* __restrict__ ctx)
{
  __shared__ float lds[8][16 * 16];   // per-wave P tile (C-layout -> A-layout)

  const int lane = threadIdx.x & 31;
  const int wib  = threadIdx.x >> 5;
  const int wid  = (int)blockIdx.x * 8 + wib;
  const int qt   = wid & 127;
  const int h    = (wid >> 7) & 7;
  const int n    = wid >> 10;
  const int t0   = qt * 16;
  const int row  = lane & 15;
  const int hi   = lane >> 4;

  const size_t bhd = ((size_t)(n * NHEAD + h)) * T_SEQ * HEAD_DIM;
  const float* Q = qws + bhd;
  const float* K = kws + bhd;
  const float* V = vws + bhd;

  // Q tile as 16 A-fragments held in registers for the whole key sweep.
  v2f qa[16];
#pragma unroll
  for (int kk = 0; kk < 16; ++kk)
    qa[kk] = *(const v2f*)(Q + (size_t)(t0 + row) * HEAD_DIM + 4 * kk + 2 * hi);

  v8f acc[4] = {};
  float mrow[8], lrow[8];
#pragma unroll
  for (int r = 0; r < 8; ++r) { mrow[r] = -3.0e38f; lrow[r] = 0.0f; }

  int s_lo = qt - 8; if (s_lo < 0)   s_lo = 0;
  int s_hi = qt + 8; if (s_hi > 127) s_hi = 127;

  for (int st = s_lo; st <= s_hi; ++st) {
    const int s0 = st * 16;

    // prefetch next key tile's K and V rows while we chew on this one
    if (st < s_hi) {
      __builtin_prefetch(K + (size_t)(s0 + 16 + row) * HEAD_DIM + 2 * hi, 0, 1);
      __builtin_prefetch(V + (size_t)(s0 + 16 + row) * HEAD_DIM + 2 * hi, 0, 1);
    }

    // S = Q * K^T  (B[k=d, j] = K[j, d]; contiguous in d -> b64 loads)
    v8f sc = {};
#pragma unroll
    for (int kk = 0; kk < 16; ++kk) {
      v2f b = *(const v2f*)(K + (size_t)(s0 + row) * HEAD_DIM + 4 * kk + 2 * hi);
      sc = wmma_f32(qa[kk], b, sc);
    }

    // scale + window mask + online softmax (half-wave shuffles, wave32)
    float tilemax[8];
#pragma unroll
    for (int r = 0; r < 8; ++r) {
      const int i = t0 + r + 8 * hi;     // query index for this VGPR row
      const int j = s0 + row;            // key index for this lane
      float v = sc[r] * 0.125f;          // HEAD_DIM^-0.5
      const bool ok = (j >= i - WIN_L) && (j <= i + WIN_R);
      v = ok ? v : -3.0e38f;
      sc[r] = v;
      float m = v;
      m = fmaxf(m, __shfl_xor(m, 1, 16));
      m = fmaxf(m, __shfl_xor(m, 2, 16));
      m = fmaxf(m, __shfl_xor(m, 4, 16));
      m = fmaxf(m, __shfl_xor(m, 8, 16));
      tilemax[r] = m;
    }

#pragma unroll
    for (int r = 0; r < 8; ++r) {
      const float newm = fmaxf(mrow[r], tilemax[r]);
      const float corr = __expf(mrow[r] - newm);
      mrow[r] = newm;
      const float pv = (sc[r] > -1.0e37f) ? __expf(sc[r] - newm) : 0.0f;
      float su = pv;
      su += __shfl_xor(su, 1, 16);
      su += __shfl_xor(su, 2, 16);
      su += __shfl_xor(su, 4, 16);
      su += __shfl_xor(su, 8, 16);
      lrow[r] = lrow[r] * corr + su;
#pragma unroll
      for (int s = 0; s < 4; ++s) acc[s][r] *= corr;
      lds[wib][(r + 8 * hi) * 16 + row] = pv;   // C-layout -> row-major in LDS
    }

    // ctx += P * V   (A = P from LDS in A-layout; B = V, strided b32 pairs)
    const float* lp = lds[wib];
#pragma unroll
    for (int kk = 0; kk < 4; ++kk) {
      v2f a = *(const v2f*)(lp + row * 16 + 4 * kk + 2 * hi);
#pragma unroll
      for (int s = 0; s < 4; ++s) {
        const int d  = s * 16 + row;
        const int k0 = s0 + 4 * kk + 2 * hi;
        v2f b;
        b.x = V[(size_t)k0 * HEAD_DIM + d];
        b.y = V[(size_t)(k0 + 1) * HEAD_DIM + d];
        acc[s] = wmma_f32(a, b, acc[s]);
      }
    }
  }

  // normalize and store ctx in [n, t, h*64+d] layout for the output GEMM
#pragma unroll
  for (int s = 0; s < 4; ++s) {
    const int d = s * 16 + row;
#pragma unroll
    for (int r = 0; r < 8; ++r) {
      const int t = t0 + r + 8 * hi;
      ctx[((size_t)n * T_SEQ + t) * D_MODEL + h * HEAD_DIM + d] =
          acc[s][r] / lrow[r];
    }
  }
}

// ---------------------------------------------------------------------------
// Kernel 3: out = ctx @ out_w^T + out_b
// ---------------------------------------------------------------------------
__global__ __launch_bounds__(256) void outproj_kernel(
    const float* __restrict__ ctx, const float* __restrict__ w,
    const float* __restrict__ bias, float* __restrict__ out)
{
  const int lane = threadIdx.x & 31;
  const int wid  = (int)((blockIdx.x * blockDim.x + threadIdx.x) >> 5);
  const int ng   = wid & 7;            // 8 column groups of 64
  const int mt   = wid >> 3;           // 512 row tiles
  const int m0   = mt * 16;
  const int e0   = ng * 64;
  const int row  = lane & 15;
  const int hi   = lane >> 4;

  v8f acc[4] = {};
  const float* crow  = ctx + (size_t)(m0 + row) * D_MODEL;
  const float* wrow0 = w + (size_t)(e0 + row) * D_MODEL;

  for (int k0 = 0; k0 < D_MODEL; k0 += 64) {
    if (k0 + 64 < D_MODEL) {
      __builtin_prefetch(crow + k0 + 64 + 2 * hi, 0, 1);
      __builtin_prefetch(wrow0 + k0 + 64 + 2 * hi, 0, 1);
    }
#pragma unroll
    for (int kk = 0; kk < 64; kk += 4) {
      const int k = k0 + kk;
      v2f a = *(const v2f*)(crow + k + 2 * hi);
#pragma unroll
      for (int s = 0; s < 4; ++s) {
        v2f b = *(const v2f*)(wrow0 + (size_t)s * 16 * D_MODEL + k + 2 * hi);
        acc[s] = wmma_f32(a, b, acc[s]);
      }
    }
  }

#pragma unroll
  for (int s = 0; s < 4; ++s) {
    const int e  = e0 + s * 16 + row;
    const float bv = bias[e];
#pragma unroll
    for (int r = 0; r < 8; ++r) {
      const int m = m0 + r + 8 * hi;
      out[(size_t)m * D_MODEL + e] = acc[s][r] + bv;
    }
  }
}

// ---------------------------------------------------------------------------
extern "C" void kernel_launch(void* const* d_in, const int* in_sizes, int n_in,
                              void* d_out, int out_size, void* d_ws, size_t ws_size,
                              hipStream_t stream) {
  (void)in_sizes; (void)n_in; (void)out_size; (void)ws_size;
  const float* x    = (const float*)d_in[0];   // [4,2048,512]
  const float* wqkv = (const float*)d_in[1];   // [1536,512]
  const float* outw = (const float*)d_in[2];   // [512,512]
  const float* outb = (const float*)d_in[3];   // [512]
  float* out = (float*)d_out;                  // [4,2048,512]
  float* ws  = (float*)d_ws;

  const size_t PER = (size_t)N_BATCH * NHEAD * T_SEQ * HEAD_DIM; // 4M floats
  float* qws  = ws;
  float* kws  = ws + PER;
  float* vws  = ws + 2 * PER;
  float* ctxb = ws + 3 * PER;                  // [N*T, 512]

  // waves: (8192/16 row tiles) * (1536/64 col groups) = 12288 -> 1536 blocks
  qkv_rope_kernel<<<1536, 256, 0, stream>>>(x, wqkv, qws, kws, vws);
  // waves: 4 * 8 * 128 = 4096 -> 512 blocks
  attn_kernel<<<512, 256, 0, stream>>>(qws, kws, vws, ctxb);
  // waves: 512 row tiles * 8 col groups = 4096 -> 512 blocks
  outproj_kernel<<<512, 256, 0, stream>>>(ctxb, outw, outb, out);
}